// TverskyLoss_15805479649593
// MI455X (gfx1250) — compile-verified
//
#include <hip/hip_runtime.h>
#include <hip/hip_bf16.h>

// Tversky loss for pred[8,19,512,512] f32, target[8,512,512] i32.
// Streaming softmax + per-(b,c) reductions; V_WMMA_F32_16X16X4_F32 does the
// lane<->lane+16 fold in full f32 precision; LDS + b128 loads finish the
// reduction deterministically inside each workgroup; f32 atomics across WGs.

#define NB   8
#define NC   19
#define HW   (512 * 512)              // 262144
#define WGS  256                      // 8 waves of 32
#define NWAVE (WGS / 32)
#define WG_PER_B 128                  // 1024 workgroups total
#define PIX_PER_WG (HW / WG_PER_B)    // 2048
#define PIX_PER_THREAD 2              // float2 loads, 2 adjacent pixels
#define ITERS (PIX_PER_WG / (WGS * PIX_PER_THREAD))   // 4
#define NACC (3 * NC)                 // 57: sum_p, tp, sum_t per class
#define ALPHA_ 0.5f
#define BETA_  0.5f
#define SMOOTH_ 1e-8f

typedef float v2f_t __attribute__((ext_vector_type(2)));
typedef float v8f_t __attribute__((ext_vector_type(8)));

// One V_WMMA_F32_16X16X4_F32 with A = ones(16x4), B = (x, 0):
// D[m,n] = sum_k B[k,n]  ->  every D element of lane l equals
// x[l&15] + x[(l&15)+16], independent of exact B/D row striping (all K rows
// are weighted 1). Returns the folded partial; caller sums lanes 0..15.
// Must execute in uniform control flow (EXEC all ones).
__device__ __forceinline__ float wmma_fold16(float x) {
    v2f_t a; a[0] = 1.0f; a[1] = 1.0f;
    v2f_t b; b[0] = x;    b[1] = 0.0f;
    v8f_t c = {};
    v8f_t d = __builtin_amdgcn_wmma_f32_16x16x4_f32(
        false, a, false, b, (short)0, c, false, false);
    return d[0];
}

__global__ void tversky_zero_kernel(float* __restrict__ acc) {
    int i = threadIdx.x;
    if (i < NACC * NB) acc[i] = 0.0f;
}

__global__ __launch_bounds__(WGS)
void tversky_main_kernel(const float* __restrict__ pred,
                         const int*   __restrict__ target,
                         float* __restrict__ acc /* [3][NB][NC] */) {
    const int wg    = blockIdx.x;
    const int b     = wg / WG_PER_B;
    const int chunk = wg % WG_PER_B;
    const int t     = threadIdx.x;
    const int lane  = t & 31;
    const int wave  = t >> 5;

    const float* __restrict__ predb = pred + (size_t)b * NC * HW;
    const int*   __restrict__ tgtb  = target + (size_t)b * HW;

    float asum[NC], atp[NC], acnt[NC];
#pragma unroll
    for (int c = 0; c < NC; ++c) { asum[c] = 0.0f; atp[c] = 0.0f; acnt[c] = 0.0f; }

    for (int i = 0; i < ITERS; ++i) {
        const int pix = chunk * PIX_PER_WG + (i * WGS + t) * PIX_PER_THREAD;
        float2 v[NC];
#pragma unroll
        for (int c = 0; c < NC; ++c)
            v[c] = *(const float2*)(predb + (size_t)c * HW + pix);  // b64, coalesced

        float mx = v[0].x, my = v[0].y;
#pragma unroll
        for (int c = 1; c < NC; ++c) { mx = fmaxf(mx, v[c].x); my = fmaxf(my, v[c].y); }

        float sx = 0.0f, sy = 0.0f;
#pragma unroll
        for (int c = 0; c < NC; ++c) {
            v[c].x = __expf(v[c].x - mx);  sx += v[c].x;
            v[c].y = __expf(v[c].y - my);  sy += v[c].y;
        }
        const float ix = 1.0f / sx;
        const float iy = 1.0f / sy;

        const int2 tg = *(const int2*)(tgtb + pix);
#pragma unroll
        for (int c = 0; c < NC; ++c) {
            const float px = v[c].x * ix;
            const float py = v[c].y * iy;
            const bool hx = (tg.x == c);
            const bool hy = (tg.y == c);
            asum[c] += px + py;
            atp[c]  += (hx ? px : 0.0f) + (hy ? py : 0.0f);
            acnt[c] += (hx ? 1.0f : 0.0f) + (hy ? 1.0f : 0.0f);
        }
    }

    // ---- reduction: WMMA fold (lane n + lane n+16) -> LDS -> 57 threads ----
    __shared__ float red[NWAVE][NACC][32];

#pragma unroll
    for (int c = 0; c < NC; ++c) {
        // uniform flow: EXEC is all ones at every WMMA
        red[wave][0 * NC + c][lane] = wmma_fold16(asum[c]);
        red[wave][1 * NC + c][lane] = wmma_fold16(atp[c]);
        red[wave][2 * NC + c][lane] = wmma_fold16(acnt[c]);
    }
    __syncthreads();

    if (t < NACC) {
        // Each folded entry is replicated in lanes l and l+16; summing lanes
        // 0..15 over all 8 waves yields the exact workgroup total.
        float s = 0.0f;
#pragma unroll
        for (int w = 0; w < NWAVE; ++w) {
#pragma unroll
            for (int q = 0; q < 4; ++q) {
                const float4 u = *(const float4*)&red[w][t][q * 4];  // ds_load_b128
                s += (u.x + u.y) + (u.z + u.w);
            }
        }
        const int kind = t / NC;
        const int c    = t % NC;
        atomicAdd(&acc[kind * (NB * NC) + b * NC + c], s);
    }
}

__global__ void tversky_final_kernel(const float* __restrict__ acc,
                                     float* __restrict__ out) {
    const int i = threadIdx.x;   // 256 threads, need NB*NC = 152
    float v = 0.0f;
    if (i < NB * NC) {
        const float sp = acc[0 * (NB * NC) + i];
        const float tp = acc[1 * (NB * NC) + i];
        const float st = acc[2 * (NB * NC) + i];
        const float fp = sp - tp;
        const float fn = st - tp;
        const float tv = (tp + SMOOTH_) / (tp + ALPHA_ * fp + BETA_ * fn + SMOOTH_);
        v = 1.0f - tv;
    }
    v += __shfl_xor(v, 1, 32);
    v += __shfl_xor(v, 2, 32);
    v += __shfl_xor(v, 4, 32);
    v += __shfl_xor(v, 8, 32);
    v += __shfl_xor(v, 16, 32);

    __shared__ float sred[8];
    if ((i & 31) == 0) sred[i >> 5] = v;
    __syncthreads();
    if (i == 0) {
        float s = 0.0f;
#pragma unroll
        for (int w = 0; w < 8; ++w) s += sred[w];
        out[0] = s / (float)(NB * NC);
    }
}

extern "C" void kernel_launch(void* const* d_in, const int* in_sizes, int n_in,
                              void* d_out, int out_size, void* d_ws, size_t ws_size,
                              hipStream_t stream) {
    (void)in_sizes; (void)n_in; (void)out_size; (void)ws_size;
    const float* pred   = (const float*)d_in[0];
    const int*   target = (const int*)d_in[1];
    float*       out    = (float*)d_out;
    float*       acc    = (float*)d_ws;      // 3*NB*NC = 456 floats of scratch

    tversky_zero_kernel<<<1, 512, 0, stream>>>(acc);
    tversky_main_kernel<<<NB * WG_PER_B, WGS, 0, stream>>>(pred, target, acc);
    tversky_final_kernel<<<1, 256, 0, stream>>>(acc, out);
}